// NeighborSearchLayer_83434034692868
// MI455X (gfx1250) — compile-verified
//
#include <hip/hip_runtime.h>
#include <stdint.h>

// CDNA5 / gfx1250, wave32 only.
typedef __attribute__((ext_vector_type(2))) float v2f;
typedef __attribute__((ext_vector_type(4))) float v4f;
typedef __attribute__((ext_vector_type(8))) float v8f;

#define RADIUS2 0.0064f  // 0.08^2
#define WAVE 32

// ---------------------------------------------------------------------------
// One V_WMMA_F32_16X16X4_F32 computes a full 16x16 tile of squared distances:
//   A[m][:] = [-2qx, -2qy, -2qz, 1]      (16x4, striped per ISA layout)
//   B[:][n] = [px, py, pz, |p|^2]^T      (4x16)
//   C[m][n] = |q_m|^2
//   D[m][n] = |q|^2 + |p|^2 - 2 q.p = |q_m - p_n|^2
//
// A-layout (32-bit 16x4): lanes 0-15 hold M=lane, VGPR0=K0, VGPR1=K1;
//                         lanes 16-31 hold M=lane-16, VGPR0=K2, VGPR1=K3.
// B mirrors A with N across lanes, so lane (hi,j) needs (px,py) [hi=0] or
// (pz,p2) [hi=1] of point n=j -- exactly one 8-byte half of a packed
// float4 (px,py,pz,p2). The prep kernel builds that packed array so the hot
// loops do ONE global_load_b64 per lane per tile, no p2 recompute, no
// cndmask selects.
// ---------------------------------------------------------------------------

// Pass 0: pack P4[m] = (px, py, pz, |p|^2).
__global__ void nsearch_prep_kernel(const float* __restrict__ P, int M,
                                    float* __restrict__ P4) {
  int i = blockIdx.x * blockDim.x + threadIdx.x;
  if (i >= M) return;
  float px = P[3 * i + 0];
  float py = P[3 * i + 1];
  float pz = P[3 * i + 2];
  v4f v;
  v.x = px; v.y = py; v.z = pz;
  v.w = px * px + py * py + pz * pz;
  *(v4f*)(P4 + 4 * i) = v;
}

__device__ __forceinline__ void make_query_frags(const float* __restrict__ Q,
                                                 int qbase, int hi, int j,
                                                 v2f& a, v8f& c0) {
  int q = qbase + j;                       // this lane's query row (for shfl)
  float qx = Q[3 * q + 0];
  float qy = Q[3 * q + 1];
  float qz = Q[3 * q + 2];
  float q2 = qx * qx + qy * qy + qz * qz;
  a.x = hi ? (-2.0f * qz) : (-2.0f * qx);
  a.y = hi ? 1.0f         : (-2.0f * qy);
  // C[m][n] = |q_m|^2 ; this lane needs rows m = r + hi*8, r = 0..7.
  // Lane L (L<16) holds q2 of query qbase+L -> broadcast via shfl.
#pragma unroll
  for (int r = 0; r < 8; ++r) c0[r] = __shfl(q2, r + (hi << 3));
}

// Pass 1: per-query neighbor counts.
__global__ void nsearch_count_kernel(const float* __restrict__ P4,
                                     const float* __restrict__ Q,
                                     int M, int Nq,
                                     int* __restrict__ counts) {
  int wave = (blockIdx.x * blockDim.x + threadIdx.x) >> 5;  // uniform per wave
  int lane = threadIdx.x & (WAVE - 1);
  int nQtiles = Nq >> 4;
  if (wave >= nQtiles) return;  // wave-uniform exit, EXEC stays all-ones

  int hi = lane >> 4;
  int j = lane & 15;
  int qbase = wave << 4;

  v2f a; v8f c0;
  make_query_frags(Q, qbase, hi, j, a, c0);

  int cnt[8];
#pragma unroll
  for (int r = 0; r < 8; ++r) cnt[r] = 0;

  const v2f* __restrict__ PB = (const v2f*)P4;  // 2 float2 halves per point
  int jj = (j << 1) + hi;                       // per-lane constant half index

  int nPtiles = M >> 4;
#pragma unroll 4
  for (int pt = 0; pt < nPtiles; ++pt) {
    v2f b = PB[(pt << 5) + jj];                 // one global_load_b64
    v8f d = __builtin_amdgcn_wmma_f32_16x16x4_f32(
        /*neg_a=*/false, a, /*neg_b=*/false, b,
        /*c_mod=*/(short)0, c0, /*reuse_a=*/false, /*reuse_b=*/false);
#pragma unroll
    for (int r = 0; r < 8; ++r) cnt[r] += (d[r] <= RADIUS2) ? 1 : 0;
  }

  // Reduce each row count across the 16-lane N-group (masks 1..8 stay in-half).
#pragma unroll
  for (int r = 0; r < 8; ++r) {
#pragma unroll
    for (int m = 8; m >= 1; m >>= 1) cnt[r] += __shfl_xor(cnt[r], m);
  }
  if (j == 0) {
#pragma unroll
    for (int r = 0; r < 8; ++r) counts[qbase + (hi << 3) + r] = cnt[r];
  }
}

// Pass 2: exclusive prefix scan -> int offsets (ws) + float row_splits (d_out).
__global__ void nsearch_scan_kernel(const int* __restrict__ counts, int Nq,
                                    int* __restrict__ offs,
                                    float* __restrict__ splits) {
  __shared__ int sh[1024];
  int t = threadIdx.x;
  const int ITEMS = (Nq + 1023) >> 10;  // 8 for Nq=8192
  int loc[16];
  int run = 0;
  for (int i = 0; i < ITEMS && i < 16; ++i) {
    int g = t * ITEMS + i;
    int v = (g < Nq) ? counts[g] : 0;
    run += v;
    loc[i] = run;  // inclusive within thread
  }
  sh[t] = run;
  __syncthreads();
  for (int off = 1; off < 1024; off <<= 1) {
    int v = (t >= off) ? sh[t - off] : 0;
    __syncthreads();
    sh[t] += v;
    __syncthreads();
  }
  int excl = sh[t] - run;  // exclusive prefix of this thread's chunk
  if (t == 0) { offs[0] = 0; splits[0] = 0.0f; }
  for (int i = 0; i < ITEMS && i < 16; ++i) {
    int g = t * ITEMS + i;
    if (g < Nq) {
      int inc = excl + loc[i];
      offs[g + 1] = inc;
      splits[g + 1] = (float)inc;  // harness output dtype is float32
    }
  }
}

// Pass 3: emit neighbor indices in row-major (query-grouped, ascending point)
// order using ballot-rank compaction.
__global__ void nsearch_fill_kernel(const float* __restrict__ P4,
                                    const float* __restrict__ Q,
                                    int M, int Nq,
                                    const int* __restrict__ offs,
                                    float* __restrict__ out_idx, int cap) {
  int wave = (blockIdx.x * blockDim.x + threadIdx.x) >> 5;
  int lane = threadIdx.x & (WAVE - 1);
  int nQtiles = Nq >> 4;
  if (wave >= nQtiles) return;

  int hi = lane >> 4;
  int j = lane & 15;
  int qbase = wave << 4;

  v2f a; v8f c0;
  make_query_frags(Q, qbase, hi, j, a, c0);

  // All 16 lanes of a group keep identical cursors for their 8 rows.
  int cur[8];
#pragma unroll
  for (int r = 0; r < 8; ++r) cur[r] = offs[qbase + (hi << 3) + r];

  const v2f* __restrict__ PB = (const v2f*)P4;
  int jj = (j << 1) + hi;

  int nPtiles = M >> 4;
  for (int pt = 0; pt < nPtiles; ++pt) {
    v2f b = PB[(pt << 5) + jj];
    v8f d = __builtin_amdgcn_wmma_f32_16x16x4_f32(
        false, a, false, b, (short)0, c0, false, false);
#pragma unroll
    for (int r = 0; r < 8; ++r) {
      bool pred = (d[r] <= RADIUS2);
      unsigned mask = (unsigned)__ballot(pred);
      unsigned half = hi ? (mask >> 16) : (mask & 0xFFFFu);
      if (pred) {
        int pos = cur[r] + __popc(half & ((1u << j) - 1u));
        if (pos < cap) out_idx[pos] = (float)((pt << 4) + j);
      }
      cur[r] += __popc(half);
    }
  }
}

extern "C" void kernel_launch(void* const* d_in, const int* in_sizes, int n_in,
                              void* d_out, int out_size, void* d_ws, size_t ws_size,
                              hipStream_t stream) {
  const float* P = (const float*)d_in[0];   // points  [M,3] f32
  const float* Q = (const float*)d_in[1];   // queries [N,3] f32
  int M  = in_sizes[0] / 3;                 // 16384
  int Nq = in_sizes[1] / 3;                 // 8192

  // ws layout: P4 first (16B aligned), then int counts / offsets.
  float* P4   = (float*)d_ws;               // [M*4]
  int* counts = (int*)(P4 + 4 * M);         // [Nq]
  int* offs   = counts + Nq;                // [Nq+1]

  float* out = (float*)d_out;
  int splitsLen = Nq + 1;
  int cap = out_size - splitsLen;           // #neighbor-index slots
  float* splits = out + cap;                // row_splits tail

  int waves = Nq >> 4;                      // one wave per 16 queries
  int threads = 256;                        // 8 wave32 per block
  int blocks = (waves * WAVE + threads - 1) / threads;

  nsearch_prep_kernel<<<(M + 255) / 256, 256, 0, stream>>>(P, M, P4);
  nsearch_count_kernel<<<blocks, threads, 0, stream>>>(P4, Q, M, Nq, counts);
  nsearch_scan_kernel<<<1, 1024, 0, stream>>>(counts, Nq, offs, splits);
  nsearch_fill_kernel<<<blocks, threads, 0, stream>>>(P4, Q, M, Nq, offs, out, cap);
}